// BaseMessageModule_66666482368913
// MI455X (gfx1250) — compile-verified
//
#include <hip/hip_runtime.h>

#define N_ATOMS 30000
#define N_PAIRS 600000
#define FDIM 128

typedef float v2f __attribute__((ext_vector_type(2)));
typedef float v8f __attribute__((ext_vector_type(8)));

// ---------------------------------------------------------------------------
// Stage 1: per-pair gather + weighted scatter-add into per-atom accumulator
//   S[n][0][f] = sum weighted               (radial)
//   S[n][1..3][f] = sum u_c * weighted      (pre-GEMM vector accumulators)
//   cnt[n] = number of pairs landing on atom n (for the per-pair bias)
// One 128-thread block per pair; one thread per feature channel.
// f32 atomics stay resident in the 192MB L2 (accumulator is 61.4 MB).
// ---------------------------------------------------------------------------
__global__ void pair_scatter_kernel(const float* __restrict__ emb,
                                    const float* __restrict__ fcut,
                                    const float* __restrict__ rij,
                                    const long long* __restrict__ pl,
                                    float* __restrict__ S,
                                    float* __restrict__ cnt)
{
    const int p = blockIdx.x;
    if (p >= N_PAIRS) return;
    const int f = threadIdx.x;                       // 0..127

    const int i = (int)pl[p];                        // receiving atom
    const int j = (int)pl[N_PAIRS + p];              // neighbor atom
    const float fc = fcut[p];
    const float rx = rij[3 * p + 0];
    const float ry = rij[3 * p + 1];
    const float rz = rij[3 * p + 2];
    const float inv = rsqrtf(rx * rx + ry * ry + rz * rz);
    const float ux = rx * inv, uy = ry * inv, uz = rz * inv;

    const float w = emb[(long)j * FDIM + f] * fc;

    float* Si = S + (long)i * (4 * FDIM);
    unsafeAtomicAdd(Si + 0 * FDIM + f, w);
    unsafeAtomicAdd(Si + 1 * FDIM + f, ux * w);
    unsafeAtomicAdd(Si + 2 * FDIM + f, uy * w);
    unsafeAtomicAdd(Si + 3 * FDIM + f, uz * w);
    if (f == 0) unsafeAtomicAdd(cnt + i, 1.0f);
}

// ---------------------------------------------------------------------------
// Stage 2: per-atom-tile WMMA GEMM + norms.
// Block = 256 threads = 8 waves; block handles one 16-atom tile; wave w owns
// output-column tile tg = w (16 of the 128 output features).
//
// Per wave: D_c[16x16] = S_c[16x128] @ W^T[128x16] for c = x,y,z via 32 steps
// of V_WMMA_F32_16X16X4_F32. The three component GEMMs share each B fragment
// (one read of W per wave) and run as three independent accumulator chains
// (ILP to hide WMMA latency, no fragment arrays -> no spills).
//
// Fragment layouts (ISA 7.12.2, 16x16x4 f32):
//   A: lane L(<16): M=L, v0=K(k0), v1=K(k0+1); lanes 16-31: K(k0+2),K(k0+3)
//   B: lane L(<16): N=L, v0=K(k0), v1=K(k0+1); lanes 16-31: K(k0+2),K(k0+3)
//   C/D vgpr r: lanes 0-15 -> (M=r, N=lane); lanes 16-31 -> (M=r+8, N=lane-16)
// ---------------------------------------------------------------------------
__global__ void atom_wmma_kernel(const float* __restrict__ S,
                                 const float* __restrict__ cnt,
                                 const float* __restrict__ W,
                                 const float* __restrict__ bvec,
                                 float* __restrict__ out)
{
    const int base = blockIdx.x * 16;                // 16-atom tile
    if (base >= N_ATOMS) return;                     // uniform

    const int tg   = threadIdx.x >> 5;               // wave id = column tile
    const int lane = threadIdx.x & 31;               // 0..31 (wave32)
    const int half = lane >> 4;                      // 0 or 1
    const int l16  = lane & 15;
    const int g    = tg * 16 + l16;                  // output feature column

    // counts for the 8 output rows this lane's accumulator registers cover
    float cnt8[8];
#pragma unroll
    for (int r = 0; r < 8; ++r)
        cnt8[r] = cnt[base + r + half * 8];

    // k-walk base pointers (K advances 4 per WMMA; this lane supplies
    // K = 4*kk + 2*half + {0,1})
    const float* wptr = W + (long)g * FDIM + half * 2;          // B frags
    const float* arow = S + (long)(base + l16) * (4 * FDIM) + half * 2;
    const float* a0p = arow + 1 * FDIM;                         // c = x
    const float* a1p = arow + 2 * FDIM;                         // c = y
    const float* a2p = arow + 3 * FDIM;                         // c = z

    v8f acc0 = {}, acc1 = {}, acc2 = {};
#pragma unroll 4
    for (int kk = 0; kk < 32; ++kk) {
        const v2f b  = *(const v2f*)(wptr + 4 * kk);
        const v2f a0 = *(const v2f*)(a0p + 4 * kk);
        const v2f a1 = *(const v2f*)(a1p + 4 * kk);
        const v2f a2 = *(const v2f*)(a2p + 4 * kk);
        acc0 = __builtin_amdgcn_wmma_f32_16x16x4_f32(false, a0, false, b,
                                                     (short)0, acc0, false, false);
        acc1 = __builtin_amdgcn_wmma_f32_16x16x4_f32(false, a1, false, b,
                                                     (short)0, acc1, false, false);
        acc2 = __builtin_amdgcn_wmma_f32_16x16x4_f32(false, a2, false, b,
                                                     (short)0, acc2, false, false);
    }

    // bias (count * b, since reference adds b once per pair), norms, store
    const float bval = bvec[g];
#pragma unroll
    for (int r = 0; r < 8; ++r) {
        const float vx = acc0[r] + cnt8[r] * bval;
        const float vy = acc1[r] + cnt8[r] * bval;
        const float vz = acc2[r] + cnt8[r] * bval;
        const int m = r + half * 8;
        out[(long)(base + m) * (2 * FDIM) + g] =
            sqrtf(vx * vx + vy * vy + vz * vz);
    }

    // radial half of output: straight copy of S slot 0; 2 atoms per wave
#pragma unroll
    for (int mm = 0; mm < 2; ++mm) {
        const int m = tg * 2 + mm;
        for (int f = lane; f < FDIM; f += 32) {
            out[(long)(base + m) * (2 * FDIM) + FDIM + f] =
                S[(long)(base + m) * (4 * FDIM) + f];
        }
    }
}

// ---------------------------------------------------------------------------
extern "C" void kernel_launch(void* const* d_in, const int* in_sizes, int n_in,
                              void* d_out, int out_size, void* d_ws, size_t ws_size,
                              hipStream_t stream)
{
    const float*     emb  = (const float*)d_in[0];      // [30000,128]
    const float*     fcut = (const float*)d_in[1];      // [600000,1]
    const float*     rij  = (const float*)d_in[2];      // [600000,3]
    const float*     W    = (const float*)d_in[3];      // [128,128]
    const float*     bvec = (const float*)d_in[4];      // [128]
    const long long* pl   = (const long long*)d_in[5];  // [2,600000] int64

    float* out = (float*)d_out;                         // [30000, 256]
    float* S   = (float*)d_ws;                          // [30000][4][128]
    float* cnt = S + (size_t)N_ATOMS * 4 * FDIM;        // [30000]

    const size_t zero_bytes =
        ((size_t)N_ATOMS * 4 * FDIM + (size_t)N_ATOMS) * sizeof(float);
    hipMemsetAsync(d_ws, 0, zero_bytes, stream);

    pair_scatter_kernel<<<N_PAIRS, 128, 0, stream>>>(emb, fcut, rij, pl, S, cnt);

    atom_wmma_kernel<<<(N_ATOMS + 15) / 16, 256, 0, stream>>>(S, cnt, W, bvec, out);
}